// PathCon_16913581212054
// MI455X (gfx1250) — compile-verified
//
#include <hip/hip_runtime.h>
#include <math.h>

typedef float v2f __attribute__((ext_vector_type(2)));
typedef float v8f __attribute__((ext_vector_type(8)));

__device__ __forceinline__ float sigmf(float x) { return 1.0f / (1.0f + __expf(-x)); }

// One 16x16 f32 output tile of  D = A(16x64) @ W(64x256) + C  using V_WMMA_F32_16X16X4_F32.
// A staged in LDS as hb[row][k] (16x64 floats).
// WT is the TRANSPOSED weight (256x64, WT[n][k] = W[k][n]) so the per-lane B fetch
// (B[kk][n], B[kk+1][n]) is one contiguous 8-byte load.
__device__ __forceinline__ v8f gemm_tile(const float* hb, const float* __restrict__ WT,
                                         int n, int ll, int lh, v8f acc) {
#pragma unroll
  for (int kc = 0; kc < 16; ++kc) {
    int kk = 4 * kc + 2 * lh;
    v2f a = *(const v2f*)(hb + ll * 64 + kk);        // A[m=ll][kk], A[ll][kk+1]
    v2f b = *(const v2f*)(WT + n * 64 + kk);         // B[kk][n],    B[kk+1][n]
    acc = __builtin_amdgcn_wmma_f32_16x16x4_f32(false, a, false, b, (short)0, acc, false, false);
  }
  return acc;
}

// ---------------------------------------------------------------------------
// Kernel 0a: project relation features through a weight matrix.
//   out(257 x N) = RF(257 x 256) @ W(256 x N)
// Exactly correct for arbitrary RF (here one-hot + zero row). Tiny; plain VALU.
__global__ void relproj_kernel(const float* __restrict__ RF, const float* __restrict__ W,
                               float* __restrict__ out, int N) {
  int idx = blockIdx.x * blockDim.x + threadIdx.x;
  if (idx >= 257 * N) return;
  int row = idx / N, col = idx % N;
  float s = 0.f;
  for (int k = 0; k < 256; ++k) s += RF[row * 256 + k] * W[k * N + col];
  out[idx] = s;
}

// Kernel 0b: transpose a 64x256 weight to 256x64 (WT[n*64+k] = W[k*256+n]).
__global__ void transpose64x256_kernel(const float* __restrict__ W, float* __restrict__ WT) {
  int idx = blockIdx.x * blockDim.x + threadIdx.x;   // 0..16383
  int n = idx >> 6, k = idx & 63;
  WT[n * 64 + k] = W[k * 256 + n];
}

// ---------------------------------------------------------------------------
// Kernel 1: hop-2 aggregation + agg0 layer.  One wave per (b,k) group.
//   nxt1[b][k][:] = relu( agg0_b + RW[rel(he)] + (1/32) * sum_j mask_j * RW[rel(e2_j)] )
// Also emits mask0[b][k] = (he != train_edge[b]).
__global__ void hop_agg_kernel(const int* __restrict__ entity_pairs,
                               const int* __restrict__ train_edges,
                               const int* __restrict__ entity2edges,
                               const int* __restrict__ edge2entities,
                               const int* __restrict__ edge2relation,
                               const float* __restrict__ agg0_b,
                               const float* __restrict__ RW,   // 257 x 64
                               float* __restrict__ nxt1,       // 8192 x 64
                               float* __restrict__ mask0) {    // 8192
  int lane = threadIdx.x & 31;
  int gid  = blockIdx.x * 8 + (threadIdx.x >> 5);   // 0..8191
  int b = gid >> 5;
  int k = gid & 31;

  int te  = train_edges[b];
  int ent = entity_pairs[b * 2 + (k >> 4)];
  int he  = entity2edges[ent * 16 + (k & 15)];
  float m0 = (he != te) ? 1.f : 0.f;
  int selfrel = edge2relation[he];
  int eA = edge2entities[he * 2 + 0];
  int eB = edge2entities[he * 2 + 1];

  // lane j owns hop-2 edge j of this group
  int ej   = entity2edges[(lane < 16 ? eA : eB) * 16 + (lane & 15)];
  float mj = (ej != te) ? (1.f / 32.f) : 0.f;       // fold mean(2x16) into mask
  int relj = edge2relation[ej];

  v2f acc;
  acc.x = agg0_b[2 * lane];
  acc.y = agg0_b[2 * lane + 1];
  {
    v2f w = *(const v2f*)(RW + selfrel * 64 + 2 * lane);   // self (unmasked, unscaled)
    acc.x += w.x; acc.y += w.y;
  }
#pragma unroll
  for (int j = 0; j < 32; ++j) {
    int   rb = __shfl(relj, j, 32);
    float mb = __shfl(mj, j, 32);
    v2f w = *(const v2f*)(RW + rb * 64 + 2 * lane);
    acc.x += mb * w.x;
    acc.y += mb * w.y;
  }
  float* o = nxt1 + gid * 64 + 2 * lane;
  o[0] = fmaxf(acc.x, 0.f);
  o[1] = fmaxf(acc.y, 0.f);
  if (lane == 0) mask0[gid] = m0;
}

// ---------------------------------------------------------------------------
// Kernel 2: ctx_h[b] = (1/32) sum_k mask0[b,k]*nxt1[b,k,:],  ctx = ctx_h @ agg1_w + agg1_b
// One wave per 16 batch rows; WMMA GEMM (16x64)@(64x256).
__global__ void ctx_kernel(const float* __restrict__ mask0,
                           const float* __restrict__ nxt1,
                           const float* __restrict__ agg1wT,   // 256 x 64 (transposed)
                           const float* __restrict__ agg1_b,   // 256
                           float* __restrict__ ctx) {          // 256 x 256
  __shared__ __align__(16) float lds2[4 * 1024];               // 4 waves x (16x64)
  int lane = threadIdx.x & 31;
  int widx = threadIdx.x >> 5;
  int wv   = blockIdx.x * 4 + widx;                            // 0..15
  int ll = lane & 15, lh = lane >> 4;
  float* hb = &lds2[widx * 1024];

  // Phase 1: masked mean -> hb[m][d]; lane owns dims 2*lane, 2*lane+1 (one b64/load)
  for (int m = 0; m < 16; ++m) {
    int b = wv * 16 + m;
    v2f s; s.x = 0.f; s.y = 0.f;
    for (int k = 0; k < 32; ++k) {
      float mk = mask0[b * 32 + k];
      v2f v = *(const v2f*)(nxt1 + (b * 32 + k) * 64 + 2 * lane);
      s.x += mk * v.x;
      s.y += mk * v.y;
    }
    hb[m * 64 + 2 * lane]     = s.x * (1.f / 32.f);
    hb[m * 64 + 2 * lane + 1] = s.y * (1.f / 32.f);
  }
  __syncthreads();

  // Phase 2: WMMA GEMM, tile-by-tile over N=256
  for (int t = 0; t < 16; ++t) {
    int n = 16 * t + ll;
    v8f acc;
    float bias = agg1_b[n];
#pragma unroll
    for (int r = 0; r < 8; ++r) acc[r] = bias;
    acc = gemm_tile(hb, agg1wT, n, ll, lh, acc);
#pragma unroll
    for (int r = 0; r < 8; ++r)
      ctx[(wv * 16 + r + 8 * lh) * 256 + n] = acc[r];
  }
}

// ---------------------------------------------------------------------------
// Kernel 3: LSTM over 2048 sequences (len 4, input via XW row-gather, hidden 64),
// snapshot at id2length, project through path_w, mean over 8 paths/batch,
// add ctx, sigmoid -> out.  16 sequences per wave; WMMA for h@Whh and last@path_w.
__global__ void path_kernel(const int* __restrict__ path_ids,
                            const int* __restrict__ id2path,
                            const int* __restrict__ id2length,
                            const float* __restrict__ XW,      // 257 x 256 (relfeat @ Wih)
                            const float* __restrict__ WhhT,    // 256 x 64 (transposed)
                            const float* __restrict__ bih,
                            const float* __restrict__ bhh,
                            const float* __restrict__ pathwT,  // 256 x 64 (transposed)
                            const float* __restrict__ path_b,  // 256
                            const float* __restrict__ ctx,     // 256 x 256
                            float* __restrict__ out) {         // 256 x 256
  __shared__ __align__(16) float ldsP[4 * 2 * 1024];           // 4 waves x double-buffered 16x64
  int lane = threadIdx.x & 31;
  int widx = threadIdx.x >> 5;
  int wv   = blockIdx.x * 4 + widx;                            // 0..127
  int ll = lane & 15, lh = lane >> 4;
  float* buf0 = &ldsP[widx * 2048];
  float* buf1 = buf0 + 1024;

  // per-lane sequence metadata: rows m = r + 8*lh
  int pid[8], lenr[8];
#pragma unroll
  for (int r = 0; r < 8; ++r) {
    int seq = wv * 16 + r + 8 * lh;
    pid[r]  = path_ids[seq];
    lenr[r] = id2length[pid[r]];
  }

  // zero h0 buffer
  for (int i = lane; i < 1024; i += 32) buf0[i] = 0.f;
  __syncthreads();

  float cst[4][8];
  float lastv[4][8];
#pragma unroll
  for (int g = 0; g < 4; ++g)
#pragma unroll
    for (int r = 0; r < 8; ++r) { cst[g][r] = 0.f; lastv[g][r] = 0.f; }

  for (int step = 0; step < 4; ++step) {
    float* cur = (step & 1) ? buf1 : buf0;
    float* nxt = (step & 1) ? buf0 : buf1;

    int xr[8];
#pragma unroll
    for (int r = 0; r < 8; ++r) xr[r] = id2path[pid[r] * 4 + step];

#pragma unroll
    for (int g = 0; g < 4; ++g) {
      int ni = 16 * g + ll;             // gate i dims  0..63
      int nf = 16 * (g + 4) + ll;       // gate f dims 64..127
      int ng = 16 * (g + 8) + ll;       // gate g dims 128..191
      int no = 16 * (g + 12) + ll;      // gate o dims 192..255
      float bi = bih[ni] + bhh[ni];
      float bf = bih[nf] + bhh[nf];
      float bg = bih[ng] + bhh[ng];
      float bo = bih[no] + bhh[no];

      v8f ai, af, ag_, ao;
#pragma unroll
      for (int r = 0; r < 8; ++r) {
        const float* xw = XW + xr[r] * 256;   // row 256 is all-zero automatically
        ai[r]  = bi + xw[ni];
        af[r]  = bf + xw[nf];
        ag_[r] = bg + xw[ng];
        ao[r]  = bo + xw[no];
      }
      ai  = gemm_tile(cur, WhhT, ni, ll, lh, ai);
      af  = gemm_tile(cur, WhhT, nf, ll, lh, af);
      ag_ = gemm_tile(cur, WhhT, ng, ll, lh, ag_);
      ao  = gemm_tile(cur, WhhT, no, ll, lh, ao);

#pragma unroll
      for (int r = 0; r < 8; ++r) {
        float cv = cst[g][r];
        cv = sigmf(af[r]) * cv + sigmf(ai[r]) * tanhf(ag_[r]);
        float hv = sigmf(ao[r]) * tanhf(cv);
        cst[g][r] = cv;
        nxt[(r + 8 * lh) * 64 + 16 * g + ll] = hv;
        lastv[g][r] = (lenr[r] == step + 1) ? hv : lastv[g][r];
      }
    }
    __syncthreads();
  }

  // stage snapshot hidden states ("last") for the output GEMM
#pragma unroll
  for (int g = 0; g < 4; ++g)
#pragma unroll
    for (int r = 0; r < 8; ++r)
      buf0[(r + 8 * lh) * 64 + 16 * g + ll] = lastv[g][r];
  __syncthreads();

  // p = last @ path_w + path_b ; path_scores = mean over 8 paths (= sum over r in-lane)
  int bidx = 2 * wv + lh;   // batch index covered by this half-wave
  for (int t = 0; t < 16; ++t) {
    int n = 16 * t + ll;
    v8f acc;
    float pb = path_b[n];
#pragma unroll
    for (int r = 0; r < 8; ++r) acc[r] = pb;
    acc = gemm_tile(buf0, pathwT, n, ll, lh, acc);
    float s = (acc[0] + acc[1] + acc[2] + acc[3] + acc[4] + acc[5] + acc[6] + acc[7]) * 0.125f;
    out[bidx * 256 + n] = sigmf(ctx[bidx * 256 + n] + s);
  }
}

// ---------------------------------------------------------------------------
extern "C" void kernel_launch(void* const* d_in, const int* in_sizes, int n_in,
                              void* d_out, int out_size, void* d_ws, size_t ws_size,
                              hipStream_t stream) {
  (void)in_sizes; (void)n_in; (void)out_size; (void)ws_size;

  const float* RF      = (const float*)d_in[0];   // 257 x 256
  const float* agg0_w  = (const float*)d_in[1];   // 256 x 64
  const float* agg0_b  = (const float*)d_in[2];   // 64
  const float* agg1_w  = (const float*)d_in[3];   // 64 x 256
  const float* agg1_b  = (const float*)d_in[4];   // 256
  const float* Wih     = (const float*)d_in[5];   // 256 x 256
  const float* Whh     = (const float*)d_in[6];   // 64 x 256
  const float* bih     = (const float*)d_in[7];   // 256
  const float* bhh     = (const float*)d_in[8];   // 256
  const float* path_w  = (const float*)d_in[9];   // 64 x 256
  const float* path_b  = (const float*)d_in[10];  // 256
  const int* entity_pairs  = (const int*)d_in[11];
  const int* train_edges   = (const int*)d_in[12];
  // d_in[13] = labels (unused: hop-0 self branch is dead in the reference)
  const int* path_ids      = (const int*)d_in[14];
  const int* entity2edges  = (const int*)d_in[15];
  const int* edge2entities = (const int*)d_in[16];
  const int* edge2relation = (const int*)d_in[17];
  const int* id2path       = (const int*)d_in[18];
  const int* id2length     = (const int*)d_in[19];

  float* out = (float*)d_out;

  // workspace layout (floats)
  float* ws     = (float*)d_ws;
  float* nxt1   = ws;                // 8192 * 64      = 524288
  float* mask0  = nxt1 + 524288;     // 8192
  float* ctx    = mask0 + 8192;      // 256 * 256      = 65536
  float* RW     = ctx + 65536;       // 257 * 64       = 16448
  float* XW     = RW + 16448;        // 257 * 256      = 65792
  float* WhhT   = XW + 65792;        // 256 * 64       = 16384
  float* agg1wT = WhhT + 16384;      // 256 * 64       = 16384
  float* pathwT = agg1wT + 16384;    // 256 * 64       = 16384

  relproj_kernel<<<(257 * 64 + 255) / 256, 256, 0, stream>>>(RF, agg0_w, RW, 64);
  relproj_kernel<<<(257 * 256 + 255) / 256, 256, 0, stream>>>(RF, Wih, XW, 256);
  transpose64x256_kernel<<<64, 256, 0, stream>>>(Whh, WhhT);
  transpose64x256_kernel<<<64, 256, 0, stream>>>(agg1_w, agg1wT);
  transpose64x256_kernel<<<64, 256, 0, stream>>>(path_w, pathwT);

  hop_agg_kernel<<<1024, 256, 0, stream>>>(entity_pairs, train_edges, entity2edges,
                                           edge2entities, edge2relation, agg0_b,
                                           RW, nxt1, mask0);

  ctx_kernel<<<4, 128, 0, stream>>>(mask0, nxt1, agg1wT, agg1_b, ctx);

  path_kernel<<<32, 128, 0, stream>>>(path_ids, id2path, id2length, XW, WhhT,
                                      bih, bhh, pathwT, path_b, ctx, out);
}